// TuckerLinear_22290880266704
// MI455X (gfx1250) — compile-verified
//
#include <hip/hip_runtime.h>

// ---------------------------------------------------------------------------
// TuckerLinear as three fused-weight GEMMs on CDNA5 WMMA (fp32, 16x16x4).
//   S   = x @ W_in          (16384 x 4096) @ (4096 x 512)
//   U   = S @ CoreT         (16384 x 512 ) @ (512  x 512)
//   out = U @ WoutT + bias  (16384 x 512 ) @ (512  x 4096)
// W_in rows are scattered by input_perm (x is read with no gather);
// WoutT absorbs per_dim_scale * alpha.
// GEMM: 128x128 block tiles, BK=32, double-buffered LDS fed by
// GLOBAL_LOAD_ASYNC_TO_LDS (ASYNCcnt) when available.
// ---------------------------------------------------------------------------

typedef float v2f __attribute__((ext_vector_type(2)));
typedef float v8f __attribute__((ext_vector_type(8)));
typedef int   v4i __attribute__((vector_size(16)));   // matches builtin proto

#define LDA 36              // A tile row stride (words): banks m*36+j cover all 64
#define ABUF (128 * LDA)    // words per A buffer
#define BBUF (32 * 128)     // words per B buffer (fragment-major [k4][col][4])

#if __has_builtin(__builtin_amdgcn_global_load_async_to_lds_b128) && \
    __has_builtin(__builtin_amdgcn_global_load_async_to_lds_b32)
#define HAVE_ASYNC_LDS 1
#else
#define HAVE_ASYNC_LDS 0
#endif

#if HAVE_ASYNC_LDS
#define GP128(p) ((__attribute__((address_space(1))) v4i*)(p))
#define LP128(p) ((__attribute__((address_space(3))) v4i*)(p))
#define GP32(p)  ((__attribute__((address_space(1))) int*)(p))
#define LP32(p)  ((__attribute__((address_space(3))) int*)(p))
#if __has_builtin(__builtin_amdgcn_s_wait_asynccnt)
#define WAIT_ASYNC_N(n) __builtin_amdgcn_s_wait_asynccnt(n)
#else
#define WAIT_ASYNC_N(n) asm volatile("s_wait_asynccnt " #n ::: "memory")
#endif
#endif

// ----------------------------- weight builders -----------------------------

__global__ __launch_bounds__(256) void build_win(
    const float* __restrict__ fi0, const float* __restrict__ fi1,
    const float* __restrict__ fi2, const int* __restrict__ perm,
    float* __restrict__ W_in) {
  int idx = blockIdx.x * 256 + threadIdx.x;       // over 4096*512
  int i = idx >> 9;                               // pre-perm feature (a,b,c)
  int k = idx & 511;                              // (p,q,r)
  int a = i >> 8, b = (i >> 4) & 15, c = i & 15;
  int p = k >> 6, q = (k >> 3) & 7, r = k & 7;
  float v = fi0[a * 8 + p] * fi1[b * 8 + q] * fi2[c * 8 + r];
  // xp[n,i] = x[n, perm[i]]  =>  s[n,k] = sum_j x[n,j] * W_in[j,k]
  // with W_in[perm[i], k] = Uin[i, k]  (perm is a bijection: every row written)
  W_in[(size_t)perm[i] * 512 + k] = v;
}

__global__ __launch_bounds__(256) void build_coret(
    const float* __restrict__ core, float* __restrict__ CoreT) {
  int idx = blockIdx.x * 256 + threadIdx.x;       // over 512*512
  int k = idx >> 9;                               // (p,q,r)
  int X = idx & 511;                              // (x,y,z)
  CoreT[idx] = core[(size_t)X * 512 + k];         // CoreT[k][X] = Cmat[X][k]
}

__global__ __launch_bounds__(256) void build_woutT(
    const float* __restrict__ fo0, const float* __restrict__ fo1,
    const float* __restrict__ fo2, const float* __restrict__ scale,
    const float* __restrict__ alpha, float* __restrict__ WoT) {
  int idx = blockIdx.x * 256 + threadIdx.x;       // over 512*4096
  int k = idx >> 12;                              // (x,y,z)
  int o = idx & 4095;                             // (X,Y,Z)
  int x = k >> 6, y = (k >> 3) & 7, z = k & 7;
  int X = o >> 8, Y = (o >> 4) & 15, Z = o & 15;
  WoT[idx] = fo0[X * 8 + x] * fo1[Y * 8 + y] * fo2[Z * 8 + z]
           * scale[o] * alpha[0];
}

// ------------------------------- GEMM kernel -------------------------------
// C[M,N] = A[M,K] @ B[K,N] (+ bias[col]).  M,N multiples of 128, K of 32.
// 256 threads = 8 wave32s; wave (rg,cg) owns a 32x64 macro-tile:
// 2 row-tiles x 4 col-tiles of 16x16 f32 WMMA, K-steps of 4.

template <bool BIAS>
__global__ __launch_bounds__(256) void gemm128x128_wmma_f32(
    const float* __restrict__ A, const float* __restrict__ Bm,
    float* __restrict__ C, const float* __restrict__ bias, int N, int K) {
  __shared__ float As[2 * ABUF];
  __shared__ float Bs[2 * BBUF];

  const int tid  = threadIdx.x;
  const int lane = tid & 31;
  const int wid  = tid >> 5;
  const int rg   = wid & 3;    // 4 row-groups of 32 rows
  const int cg   = wid >> 2;   // 2 col-groups of 64 cols
  const int row0 = blockIdx.y * 128;
  const int col0 = blockIdx.x * 128;

  // A-frag layout (32b A 16x4): lane m=L&15; lanes<16 -> K=0,1; lanes>=16 -> K=2,3
  const int m  = lane & 15;
  const int kk = (lane >> 4) << 1;
  const int hi = lane >> 4;

  v8f acc[2][4] = {};

  // ---- stage one BK=32 chunk into LDS buffer `buf` -------------------------
  auto stage = [&](int kb, int buf) {
    float* Ad = &As[buf * ABUF];
    float* Bd = &Bs[buf * BBUF];
#if HAVE_ASYNC_LDS
    // A tile: 128 rows x 32 K -> 4 async b128 per thread
#pragma unroll
    for (int i = 0; i < 4; ++i) {
      int idx = tid + i * 256;
      int r = idx >> 3, c4 = idx & 7;
      __builtin_amdgcn_global_load_async_to_lds_b128(
          GP128(&A[(size_t)(row0 + r) * K + kb + (c4 << 2)]),
          LP128(&Ad[r * LDA + (c4 << 2)]), 0, 0);
    }
    // B tile: fragment-major Bd[(k4*128 + col)*4 + j] -> 16 async b32 / thread
#pragma unroll
    for (int i = 0; i < 4; ++i) {
      int idx = tid + i * 256;
      int col = idx & 127, kc = idx >> 7;
#pragma unroll
      for (int j = 0; j < 4; ++j) {
        __builtin_amdgcn_global_load_async_to_lds_b32(
            GP32(&Bm[(size_t)(kb + (kc << 2) + j) * N + col0 + col]),
            LP32(&Bd[(((kc << 7) + col) << 2) + j]), 0, 0);
      }
    }
#else
    // sync fallback (proven path)
#pragma unroll
    for (int i = 0; i < 4; ++i) {
      int idx = tid + i * 256;
      int r = idx >> 3, c4 = idx & 7;
      const float4 v =
          *(const float4*)&A[(size_t)(row0 + r) * K + kb + (c4 << 2)];
      *(float4*)&Ad[r * LDA + (c4 << 2)] = v;
    }
#pragma unroll
    for (int i = 0; i < 4; ++i) {
      int idx = tid + i * 256;
      int col = idx & 127, kc = idx >> 7;
      float4 v;
      v.x = Bm[(size_t)(kb + (kc << 2) + 0) * N + col0 + col];
      v.y = Bm[(size_t)(kb + (kc << 2) + 1) * N + col0 + col];
      v.z = Bm[(size_t)(kb + (kc << 2) + 2) * N + col0 + col];
      v.w = Bm[(size_t)(kb + (kc << 2) + 3) * N + col0 + col];
      *(float4*)&Bd[((kc << 7) + col) << 2] = v;
    }
#endif
  };

  // ---- consume one chunk ---------------------------------------------------
  auto compute = [&](int buf) {
    const float* Ab = &As[buf * ABUF];
    const float* Bb = &Bs[buf * BBUF];
#pragma unroll
    for (int k4 = 0; k4 < 8; ++k4) {
      v2f af[2];
#pragma unroll
      for (int rt = 0; rt < 2; ++rt)
        af[rt] =
            *(const v2f*)&Ab[(rg * 32 + rt * 16 + m) * LDA + (k4 << 2) + kk];
      v2f bf[4];
#pragma unroll
      for (int ct = 0; ct < 4; ++ct) {
        int cb = cg * 64 + ct * 16 + m;  // n = lane&15
        bf[ct] = *(const v2f*)&Bb[((((k4 << 7) + cb)) << 2) + kk];
      }
#pragma unroll
      for (int rt = 0; rt < 2; ++rt)
#pragma unroll
        for (int ct = 0; ct < 4; ++ct)
          acc[rt][ct] = __builtin_amdgcn_wmma_f32_16x16x4_f32(
              false, af[rt], false, bf[ct], (short)0, acc[rt][ct],
              false, false);
    }
  };

  // ---- double-buffered main loop ------------------------------------------
  const int nc = K >> 5;
  stage(0, 0);
  for (int c = 0; c < nc; ++c) {
    const int cur = c & 1;
    if (c + 1 < nc) {
      stage((c + 1) << 5, cur ^ 1);
#if HAVE_ASYNC_LDS
      WAIT_ASYNC_N(20);  // 20 async ops/thread in flight for chunk c+1
#endif
    } else {
#if HAVE_ASYNC_LDS
      WAIT_ASYNC_N(0);
#endif
    }
    __syncthreads();
    compute(cur);
    if (c + 1 < nc) __syncthreads();  // buffer reuse fence
  }

  // ---- epilogue: C/D 16x16 layout -> col = L&15, row = vgpr + 8*(L>>4) ----
#pragma unroll
  for (int rt = 0; rt < 2; ++rt) {
#pragma unroll
    for (int ct = 0; ct < 4; ++ct) {
      int col = col0 + cg * 64 + ct * 16 + m;
      float badd = BIAS ? bias[col] : 0.0f;
#pragma unroll
      for (int r = 0; r < 8; ++r) {
        int row = row0 + rg * 32 + rt * 16 + r + 8 * hi;
        C[(size_t)row * N + col] = acc[rt][ct][r] + badd;
      }
    }
  }
}

// ------------------------------- launcher ----------------------------------

extern "C" void kernel_launch(void* const* d_in, const int* in_sizes, int n_in,
                              void* d_out, int out_size, void* d_ws,
                              size_t ws_size, hipStream_t stream) {
  const float* x     = (const float*)d_in[0];
  const float* core  = (const float*)d_in[1];
  const float* fi0   = (const float*)d_in[2];
  const float* fi1   = (const float*)d_in[3];
  const float* fi2   = (const float*)d_in[4];
  const float* fo0   = (const float*)d_in[5];
  const float* fo1   = (const float*)d_in[6];
  const float* fo2   = (const float*)d_in[7];
  const float* pds   = (const float*)d_in[8];
  const float* alpha = (const float*)d_in[9];
  const float* bias  = (const float*)d_in[10];
  const int*   perm  = (const int*)d_in[11];

  float* ws    = (float*)d_ws;
  float* W_in  = ws;                          //  4096*512  ( 8 MB)
  float* CoreT = W_in + 4096 * 512;           //   512*512  ( 1 MB)
  float* WoT   = CoreT + 512 * 512;           //  512*4096  ( 8 MB)
  float* S     = WoT + 512 * 4096;            // 16384*512  (32 MB)
  float* U     = S + 16384 * 512;             // 16384*512  (32 MB)

  build_win  <<<(4096 * 512) / 256, 256, 0, stream>>>(fi0, fi1, fi2, perm, W_in);
  build_coret<<<( 512 * 512) / 256, 256, 0, stream>>>(core, CoreT);
  build_woutT<<<(512 * 4096) / 256, 256, 0, stream>>>(fo0, fo1, fo2, pds, alpha, WoT);

  const int M = 16384;  // 4 * 4096 rows
  gemm128x128_wmma_f32<false>
      <<<dim3(512 / 128, M / 128), 256, 0, stream>>>(x, W_in, S, nullptr, 512, 4096);
  gemm128x128_wmma_f32<false>
      <<<dim3(512 / 128, M / 128), 256, 0, stream>>>(S, CoreT, U, nullptr, 512, 512);
  gemm128x128_wmma_f32<true>
      <<<dim3(4096 / 128, M / 128), 256, 0, stream>>>(U, WoT, (float*)d_out, bias, 4096, 512);
}